// QuantizationLoss_7138235646171
// MI455X (gfx1250) — compile-verified
//
#include <hip/hip_runtime.h>

typedef float v2f __attribute__((ext_vector_type(2)));
typedef float v4f __attribute__((ext_vector_type(4)));
typedef float v8f __attribute__((ext_vector_type(8)));

// Closed-form distance to nearest quantization level i/127.5 - 1 (i in [0,255]),
// clamped by the reference's init value of 1.0.
__device__ __forceinline__ float qdist(float x) {
    float t = fmaf(x, 127.5f, 127.5f);        // (x + 1) * 127.5
    float i = __builtin_rintf(t);             // v_rndne_f32; ties are equidistant
    i = fminf(fmaxf(i, 0.0f), 255.0f);
    float lv = fmaf(i, (1.0f / 127.5f), -1.0f);
    return fminf(fabsf(x - lv), 1.0f);
}

__device__ __forceinline__ float qdist4(v4f v) {
    return (qdist(v.x) + qdist(v.y)) + (qdist(v.z) + qdist(v.w));
}

// Reduce 256 per-thread partials to one float with V_WMMA_F32_16X16X4_F32:
//   P = 16x16 matrix of partials in LDS.
//   For j = 0..3:  C += P[:, 4j:4j+4] (16x4)  x  ones (4x16)
//   => C[m, n] = row-sum_m for every n.  Sum the 8 C VGPRs per lane, then
//   combine rows 0-7 (lanes 0-15) with rows 8-15 (lanes 16-31) via shfl_xor.
// Valid in wave 0; read the result from threadIdx.x == 0.
__device__ __forceinline__ float block_sum_256_wmma(float partial, float* sdata) {
    sdata[threadIdx.x] = partial;
    __syncthreads();
    float total = 0.0f;
    if (threadIdx.x < 32) {                   // wave 0 only: EXEC all-ones inside
        const int lane = threadIdx.x;
        const int m  = lane & 15;             // A-matrix row
        const int kb = (lane >> 4) << 1;      // lanes 0-15 -> K 0,1 ; lanes 16-31 -> K 2,3
        v2f b; b.x = 1.0f; b.y = 1.0f;        // all-ones B (4x16)
        v8f c = {};
        #pragma unroll
        for (int j = 0; j < 4; ++j) {
            v2f a;
            a.x = sdata[m * 16 + 4 * j + kb + 0];
            a.y = sdata[m * 16 + 4 * j + kb + 1];
            // (neg_a, A, neg_b, B, c_mod, C, reuse_a, reuse_b)
            c = __builtin_amdgcn_wmma_f32_16x16x4_f32(
                    false, a, false, b, (short)0, c, false, false);
        }
        float s = 0.0f;
        #pragma unroll
        for (int r = 0; r < 8; ++r) s += c[r];
        total = s + __shfl_xor(s, 16, 32);    // rows 0-7 + rows 8-15
    }
    return total;
}

__global__ __launch_bounds__(256)
void qloss_partial(const float* __restrict__ x, float* __restrict__ blocksum,
                   long long nvec, long long n) {
    __shared__ float sdata[256];
    const long long stride = (long long)gridDim.x * blockDim.x;
    const long long tid    = (long long)blockIdx.x * blockDim.x + threadIdx.x;
    const v4f* __restrict__ xv = (const v4f*)x;

    // 4-deep unroll over the grid stride: 4 independent global_load_b128 in
    // flight per wave (MLP for the HBM stream) + 4 independent accumulators.
    float a0 = 0.0f, a1 = 0.0f, a2 = 0.0f, a3 = 0.0f;
    long long i = tid;
    for (; i + 3 * stride < nvec; i += 4 * stride) {
        v4f v0 = __builtin_nontemporal_load(xv + i);
        v4f v1 = __builtin_nontemporal_load(xv + i + stride);
        v4f v2 = __builtin_nontemporal_load(xv + i + 2 * stride);
        v4f v3 = __builtin_nontemporal_load(xv + i + 3 * stride);
        a0 += qdist4(v0);
        a1 += qdist4(v1);
        a2 += qdist4(v2);
        a3 += qdist4(v3);
    }
    for (; i < nvec; i += stride) {
        a0 += qdist4(__builtin_nontemporal_load(xv + i));
    }
    float acc = (a0 + a1) + (a2 + a3);

    if (blockIdx.x == 0 && threadIdx.x == 0) {        // scalar tail (n % 4)
        for (long long j = nvec * 4; j < n; ++j) acc += qdist(x[j]);
    }

    float total = block_sum_256_wmma(acc, sdata);
    if (threadIdx.x == 0) blocksum[blockIdx.x] = total;
}

__global__ __launch_bounds__(256)
void qloss_final(const float* __restrict__ blocksum, float* __restrict__ out,
                 int nblocks) {
    __shared__ float sdata[256];
    float acc = 0.0f;
    for (int i = threadIdx.x; i < nblocks; i += 256) acc += blocksum[i];
    float total = block_sum_256_wmma(acc, sdata);
    if (threadIdx.x == 0) out[0] = total;
}

extern "C" void kernel_launch(void* const* d_in, const int* in_sizes, int n_in,
                              void* d_out, int out_size, void* d_ws, size_t ws_size,
                              hipStream_t stream) {
    const float* x = (const float*)d_in[0];
    float* out = (float*)d_out;
    const long long n    = (long long)in_sizes[0];
    const long long nvec = n >> 2;

    int blocks = 2048;
    if (ws_size < (size_t)blocks * sizeof(float)) {
        // scratch too small for the two-stage plan: degrade gracefully
        blocks = (int)(ws_size / sizeof(float));
        if (blocks < 1) {
            // no scratch at all: single block writes the final sum directly
            qloss_partial<<<1, 256, 0, stream>>>(x, out, nvec, n);
            return;
        }
    }

    float* blocksum = (float*)d_ws;
    qloss_partial<<<blocks, 256, 0, stream>>>(x, blocksum, nvec, n);
    qloss_final<<<1, 256, 0, stream>>>(blocksum, out, blocks);
}